// LTCCell_48326972014684
// MI455X (gfx1250) — compile-verified
//
#include <hip/hip_runtime.h>
#include <hip/hip_bf16.h>
#include <stdint.h>

#define S_DIM 32
#define U_DIM 64
#define M_DIM 8
#define N_UNFOLDS 6
#define EPS_C 1e-8f
#define LOG2E 1.4426950408889634f

// ---------------- prep: pack parameters into d_ws ----------------
// d_ws layout (bytes):
//   [0      , 32768) : P1  : 4096 float2  {-sigma*log2e, mu*sigma*log2e}   [i*64+j]
//   [32768  , 49152) : Pw  : 4096 float   {w*mask*erev}                    [i*64+j]
//   [49152  , 65536) : SP1 : 2048 float2  {-ssig*log2e, smu*ssig*log2e}    [s*64+j]
//   [65536  , 73728) : SPw : 2048 float   {sw*smask*serev}                 [s*64+j]
__global__ void ltc_prep(const float* __restrict__ sigma, const float* __restrict__ mu,
                         const float* __restrict__ w, const float* __restrict__ erev,
                         const float* __restrict__ mask,
                         const float* __restrict__ smu, const float* __restrict__ ssig,
                         const float* __restrict__ sw, const float* __restrict__ serev,
                         const float* __restrict__ smask,
                         float2* __restrict__ P1, float* __restrict__ Pw,
                         float2* __restrict__ SP1, float* __restrict__ SPw) {
  int idx = blockIdx.x * blockDim.x + threadIdx.x;   // 0..4095
  if (idx < U_DIM * U_DIM) {
    float sg = sigma[idx], m = mu[idx];
    P1[idx] = make_float2(-sg * LOG2E, m * sg * LOG2E);
    Pw[idx] = w[idx] * mask[idx] * erev[idx];
  }
  if (idx < S_DIM * U_DIM) {
    float sg = ssig[idx], m = smu[idx];
    SP1[idx] = make_float2(-sg * LOG2E, m * sg * LOG2E);
    SPw[idx] = sw[idx] * smask[idx] * serev[idx];
  }
}

// ---------------- main LTC kernel ----------------
// 256 threads: j = tid & 63 (target unit), g = tid >> 6 (batch sub-group).
// Each thread handles 4 batch elements => workgroup covers 16 batch elements.
__global__ __launch_bounds__(256) void ltc_main(
    const float* __restrict__ inputs, const float* __restrict__ states,
    const float* __restrict__ gleak, const float* __restrict__ vleak,
    const float* __restrict__ cm,
    const float* __restrict__ input_w, const float* __restrict__ input_b,
    const float* __restrict__ output_w, const float* __restrict__ output_b,
    const float* __restrict__ kst, const float* __restrict__ krel,
    const float* __restrict__ kinf, const float* __restrict__ kpump,
    const float* __restrict__ gParams,   // P1 + Pw contiguous (48 KB)
    const float2* __restrict__ gSP1, const float* __restrict__ gSPw,
    float* __restrict__ out, float* __restrict__ ns) {
  // 48 KB packed recurrent params, async-DMA'd from global.
  __shared__ __align__(16) float sRaw[12288];
  __shared__ float4 sV[4 * U_DIM];       // v[b] packed 4-wide per group  (4 KB)
  __shared__ float  sX[16 * S_DIM];      // mapped sensory inputs        (2 KB)

  const int tid = threadIdx.x;
  const int j = tid & 63;
  const int g = tid >> 6;
  const int wgBase = blockIdx.x * 16;
  const int b0 = wgBase + g * 4;

  // ---- 1) kick off async param staging (CDNA5 async-to-LDS path) ----
  {
    unsigned ldsBase = (unsigned)(uintptr_t)&sRaw[0];
    #pragma unroll
    for (int c = 0; c < 12; ++c) {                 // 12 * 256 * 16B = 48 KB
      unsigned off = (unsigned)(tid + c * 256) * 16u;
      unsigned lds = ldsBase + off;
      asm volatile("global_load_async_to_lds_b128 %0, %1, %2"
                   :: "v"(lds), "v"(off), "s"(gParams) : "memory");
    }
  }

  // ---- 2) initial v into registers + LDS (float4 packed per (g,i)) ----
  float vc[4];
  #pragma unroll
  for (int k = 0; k < 4; ++k) vc[k] = states[(size_t)(b0 + k) * (U_DIM + 1) + j];
  sV[g * U_DIM + j] = make_float4(vc[0], vc[1], vc[2], vc[3]);

  // ---- 3) stage mapped sensory inputs x = in*iw+ib into LDS ----
  #pragma unroll
  for (int r = 0; r < 2; ++r) {
    int t2 = tid + r * 256;                 // 0..511
    int bl = t2 >> 5, s = t2 & 31;
    sX[bl * S_DIM + s] =
        fmaf(inputs[(size_t)(wgBase + bl) * S_DIM + s], input_w[s], input_b[s]);
  }
  __syncthreads();   // sX, sV visible (async DMA still in flight)

  // per-thread (per target unit j) scalars
  const float cmt = cm[j] * (float)N_UNFOLDS;   // cm / (DT/UNFOLDS), DT=1
  const float gl  = gleak[j];
  const float gv  = gl * vleak[j];

  // ---- 4) sensory pass (overlapped with async DMA; params from L2) ----
  float sn[4] = {0.f, 0.f, 0.f, 0.f}, sd[4] = {0.f, 0.f, 0.f, 0.f};
  #pragma unroll 8
  for (int s = 0; s < S_DIM; ++s) {
    float2 p1 = gSP1[s * U_DIM + j];
    float  we = gSPw[s * U_DIM + j];
    float  wa = fabsf(we);
    #pragma unroll
    for (int k = 0; k < 4; ++k) {
      float x = sX[(g * 4 + k) * S_DIM + s];
      float t = fmaf(x, p1.x, p1.y);                       // -z*log2e
      float e = __builtin_amdgcn_exp2f(t);                 // v_exp_f32
      float sig = __builtin_amdgcn_rcpf(1.0f + e);         // v_rcp_f32
      sn[k] = fmaf(we, sig, sn[k]);
      sd[k] = fmaf(wa, sig, sd[k]);
    }
  }

  // ---- wait for async param DMA, then make LDS visible WG-wide ----
  asm volatile("s_wait_asynccnt 0" ::: "memory");
  __syncthreads();

  const float2* sP1 = reinterpret_cast<const float2*>(sRaw);   // 4096 float2
  const float*  sPw = sRaw + 8192;                             // 4096 float

  // ---- 5) six recurrent unfolds ----
  #pragma unroll 1
  for (int u = 0; u < N_UNFOLDS; ++u) {
    float num[4], den[4];
    #pragma unroll
    for (int k = 0; k < 4; ++k) { num[k] = sn[k]; den[k] = sd[k]; }

    #pragma unroll 8
    for (int i = 0; i < U_DIM; ++i) {
      float2 p1 = sP1[i * U_DIM + j];          // ds_load_b64, conflict-free
      float  we = sPw[i * U_DIM + j];          // ds_load_b32, conflict-free
      float  wa = fabsf(we);
      float4 vi = sV[g * U_DIM + i];           // ds_load_b128, wave broadcast
      float vv[4] = {vi.x, vi.y, vi.z, vi.w};
      #pragma unroll
      for (int k = 0; k < 4; ++k) {
        float t = fmaf(vv[k], p1.x, p1.y);
        float e = __builtin_amdgcn_exp2f(t);
        float sig = __builtin_amdgcn_rcpf(1.0f + e);
        num[k] = fmaf(we, sig, num[k]);
        den[k] = fmaf(wa, sig, den[k]);
      }
    }

    float vn[4];
    #pragma unroll
    for (int k = 0; k < 4; ++k)
      vn[k] = (fmaf(cmt, vc[k], gv) + num[k]) / (cmt + gl + den[k] + EPS_C);

    __syncthreads();                                   // all reads of sV done
    sV[g * U_DIM + j] = make_float4(vn[0], vn[1], vn[2], vn[3]);
    __syncthreads();                                   // writes visible
    #pragma unroll
    for (int k = 0; k < 4; ++k) vc[k] = vn[k];
  }

  // ---- 6) outputs ----
  #pragma unroll
  for (int k = 0; k < 4; ++k)
    ns[(size_t)(b0 + k) * (U_DIM + 1) + j] = vc[k];    // coalesced over j

  if (j < M_DIM) {
    float ow = output_w[j], ob = output_b[j];
    #pragma unroll
    for (int k = 0; k < 4; ++k)
      out[(size_t)(b0 + k) * M_DIM + j] = fmaf(vc[k], ow, ob);
  }

  if (j == 0) {   // karst scalar state, 4 batch elements per group
    float storage = kst[0], release = krel[0], infil = kinf[0], pump = kpump[0];
    #pragma unroll
    for (int k = 0; k < 4; ++k) {
      int b = b0 + k;
      float S5  = states[(size_t)b * (U_DIM + 1) + U_DIM];
      float Pp  = inputs[(size_t)b * S_DIM + 0];
      float GWE = inputs[(size_t)b * S_DIM + 2];
      float a1 = (S5 / storage) * 10.0f;
      float a2 = S5 / (storage * 0.1f);
      float s1 = __builtin_amdgcn_rcpf(1.0f + __builtin_amdgcn_exp2f(-a1 * LOG2E));
      float s2 = __builtin_amdgcn_rcpf(1.0f + __builtin_amdgcn_exp2f(-a2 * LOG2E));
      ns[(size_t)b * (U_DIM + 1) + U_DIM] =
          S5 + Pp * infil - S5 * release * s1 - GWE * pump * s2;
    }
  }
}

// ---------------- host launcher ----------------
extern "C" void kernel_launch(void* const* d_in, const int* in_sizes, int n_in,
                              void* d_out, int out_size, void* d_ws, size_t ws_size,
                              hipStream_t stream) {
  const float* inputs  = (const float*)d_in[0];
  const float* states  = (const float*)d_in[1];
  const float* gleak   = (const float*)d_in[2];
  const float* vleak   = (const float*)d_in[3];
  const float* cmv     = (const float*)d_in[4];
  const float* sigma   = (const float*)d_in[5];
  const float* mu      = (const float*)d_in[6];
  const float* w       = (const float*)d_in[7];
  const float* erev    = (const float*)d_in[8];
  const float* smu     = (const float*)d_in[9];
  const float* ssig    = (const float*)d_in[10];
  const float* sw      = (const float*)d_in[11];
  const float* serev   = (const float*)d_in[12];
  const float* input_w = (const float*)d_in[13];
  const float* input_b = (const float*)d_in[14];
  const float* output_w= (const float*)d_in[15];
  const float* output_b= (const float*)d_in[16];
  const float* kst     = (const float*)d_in[17];
  const float* krel    = (const float*)d_in[18];
  const float* kinf    = (const float*)d_in[19];
  const float* kpump   = (const float*)d_in[20];
  const float* mask    = (const float*)d_in[21];
  const float* smask   = (const float*)d_in[22];

  char* ws = (char*)d_ws;
  float2* P1  = (float2*)(ws + 0);
  float*  Pw  = (float*)(ws + 32768);
  float2* SP1 = (float2*)(ws + 49152);
  float*  SPw = (float*)(ws + 65536);

  const int Bv = in_sizes[0] / S_DIM;   // 8192

  ltc_prep<<<16, 256, 0, stream>>>(sigma, mu, w, erev, mask,
                                   smu, ssig, sw, serev, smask,
                                   P1, Pw, SP1, SPw);

  float* out = (float*)d_out;
  float* nsp = out + (size_t)Bv * M_DIM;
  ltc_main<<<Bv / 16, 256, 0, stream>>>(inputs, states, gleak, vleak, cmv,
                                        input_w, input_b, output_w, output_b,
                                        kst, krel, kinf, kpump,
                                        (const float*)P1, SP1, SPw, out, nsp);
}